// ClassifyMCLoss_90726889161276
// MI455X (gfx1250) — compile-verified
//
#include <hip/hip_runtime.h>
#include <hip/hip_bf16.h>
#include <math.h>
#include <stdint.h>

#define FG_STCH 1
#define CLS_POS_IOU_THR 0.2f
#define ENTITY_PROB_THR 0.1f
#define REMOVE_THR 0.9f
#define THETA 0.1f

// ---------------------------------------------------------------------------
// Kernel 1: per-(b,ch) max over H*W = 65536 floats. 1024 rows, 256 MB total.
// HBM-bound (~11 us floor @ 23.3 TB/s). Data streamed via CDNA5 async
// global->LDS b128 copies (ASYNCcnt), 8-deep per-lane pipeline so the chip
// keeps ~32 MB of requests in flight (latency*BW coverage), no barriers in
// the steady-state loop (each lane consumes only its own 16B slots).
// ---------------------------------------------------------------------------
__global__ __launch_bounds__(256) void maxprob_kernel(const float* __restrict__ src,
                                                      float* __restrict__ out) {
  const int row = blockIdx.x;                  // b*64 + c  (1024 rows)
  const int t   = threadIdx.x;                 // 0..255
  const float4* g = (const float4*)(src + (size_t)row * 65536);  // 16384 float4/row

  constexpr int DEPTH = 8;                     // async pipeline depth per lane
  // DEPTH slots x (256 threads x 16B) = 32 KB staging, slot-major so each
  // slot's 32-lane ds_load_b128 reads 512B contiguous LDS (conflict-friendly).
  __shared__ __align__(16) float stage[DEPTH * 256 * 4];
  __shared__ float red[256];

  // LDS byte offsets of this lane's slots (low 32 bits of the flat shared
  // address == LDS offset per CDNA5 flat->LDS mapping).
  unsigned ldsoff[DEPTH];
  #pragma unroll
  for (int s = 0; s < DEPTH; ++s)
    ldsoff[s] = (unsigned)(uintptr_t)(&stage[s * 1024 + t * 4]);

  // Prologue: issue chunks 0..DEPTH-2 into slots 0..DEPTH-2.
  #pragma unroll
  for (int s = 0; s < DEPTH - 1; ++s) {
    uint64_t ga = (uint64_t)(uintptr_t)(g + s * 256 + t);
    asm volatile("global_load_async_to_lds_b128 %0, %1, off"
                 :: "v"(ldsoff[s]), "v"(ga) : "memory");
  }

  float vmax = -3.402823466e+38f;
  for (int k = 0; k < 64; ++k) {
    const int slot = k & (DEPTH - 1);
    if (k + DEPTH - 1 < 64) {
      // Keep DEPTH chunks in flight; after this wait, <= DEPTH-1 outstanding
      // => chunk k is complete (async loads complete in order within a wave).
      uint64_t ga = (uint64_t)(uintptr_t)(g + (k + DEPTH - 1) * 256 + t);
      asm volatile("global_load_async_to_lds_b128 %0, %1, off"
                   :: "v"(ldsoff[(k + DEPTH - 1) & (DEPTH - 1)]), "v"(ga)
                   : "memory");
      asm volatile("s_wait_asynccnt 7" ::: "memory");
    } else {
      asm volatile("s_wait_asynccnt 0" ::: "memory");
    }
    const float4 v = *(const float4*)(&stage[slot * 1024 + t * 4]);
    vmax = fmaxf(vmax, fmaxf(fmaxf(v.x, v.y), fmaxf(v.z, v.w)));
  }

  // Block tree-reduce the 256 partial maxima.
  red[t] = vmax;
  __syncthreads();
  for (int s = 128; s > 0; s >>= 1) {
    if (t < s) red[t] = fmaxf(red[t], red[t + s]);
    __syncthreads();
  }
  if (t == 0) out[row] = red[0];
}

// ---------------------------------------------------------------------------
// Kernel 2: gather + weights + smooth-L1 + log-softmax(81) + weighted sums.
// One thread per (b, j); 16*64 = 1024 threads in a single block (32 waves).
// w = ones(C) so w[cls] = w[0] = 1.
// ---------------------------------------------------------------------------
__global__ __launch_bounds__(1024) void loss_kernel(
    const float* __restrict__ cls_logits,   // (16,64,81)
    const float* __restrict__ iou_scores,   // (16,64,1)
    const int*   __restrict__ target_ids,   // (16,64)
    const int*   __restrict__ map_indices,  // (16,2,64)
    const float* __restrict__ map_ious,     // (16,64)
    const float* __restrict__ rand_vals,    // (16,64)
    const float* __restrict__ max_prob,     // (16,64) from kernel 1
    float* __restrict__ out) {              // [iou_loss, cls_loss]
  constexpr int CH = 64, C = 81;
  const int t = threadIdx.x;
  const int b = t >> 6;
  const int j = t & 63;

  float s_w = 0.f, s_iou = 0.f, s_ce = 0.f;
  if (j >= FG_STCH) {
    const int pj  = map_indices[(b * 2 + 0) * CH + j];
    const int gj  = map_indices[(b * 2 + 1) * CH + j];
    const int tid = target_ids[b * CH + gj];
    int cls = tid - FG_STCH + 1;
    if (cls < 0) cls = 0;
    if (cls > C - 1) cls = C - 1;

    const float iou = map_ious[b * CH + j];
    const float mp  = max_prob[b * CH + pj];
    const float rnd = rand_vals[b * CH + j];
    const bool remove = (mp < ENTITY_PROB_THR) && (rnd < REMOVE_THR);
    const float wght = remove ? 0.f : ((iou < CLS_POS_IOU_THR) ? 1.f : 2.f);

    // adjust_smooth_l1_loss(|pred_iou - iou|)
    const float piou = iou_scores[b * CH + pj];
    const float y = fabsf(piou - iou);
    const float il = (y < THETA) ? (y * y * (1.f / (2.f * THETA)))
                                 : (y - THETA + 0.5f * THETA);

    // log_softmax over 81 logits of row (b, pj); ce = -logp[cls]  (w[cls]=1)
    const float* lg = cls_logits + (size_t)(b * CH + pj) * C;
    float mx = -3.402823466e+38f;
    for (int c = 0; c < C; ++c) mx = fmaxf(mx, lg[c]);
    float se = 0.f;
    for (int c = 0; c < C; ++c) se += expf(lg[c] - mx);
    const float ce = -(lg[cls] - mx - logf(se));

    s_w = wght;
    s_iou = il * wght;
    s_ce = ce * wght;
  }

  // wave32 reduction, then cross-wave via LDS.
  #pragma unroll
  for (int off = 16; off > 0; off >>= 1) {
    s_w   += __shfl_down(s_w,   off, 32);
    s_iou += __shfl_down(s_iou, off, 32);
    s_ce  += __shfl_down(s_ce,  off, 32);
  }
  __shared__ float rw[32], ri[32], rc[32];
  const int wave = t >> 5, lane = t & 31;
  if (lane == 0) { rw[wave] = s_w; ri[wave] = s_iou; rc[wave] = s_ce; }
  __syncthreads();
  if (wave == 0) {
    float a = rw[lane], bsum = ri[lane], csum = rc[lane];
    #pragma unroll
    for (int off = 16; off > 0; off >>= 1) {
      a    += __shfl_down(a,    off, 32);
      bsum += __shfl_down(bsum, off, 32);
      csum += __shfl_down(csum, off, 32);
    }
    if (lane == 0) {
      const float wsum = a + 0.0001f;
      out[0] = bsum / wsum;  // iou_loss
      out[1] = csum / wsum;  // cls_loss
    }
  }
}

extern "C" void kernel_launch(void* const* d_in, const int* in_sizes, int n_in,
                              void* d_out, int out_size, void* d_ws, size_t ws_size,
                              hipStream_t stream) {
  const float* cls_logits     = (const float*)d_in[0];
  const float* iou_scores     = (const float*)d_in[1];
  const int*   target_ids     = (const int*)  d_in[2];
  const int*   map_indices    = (const int*)  d_in[3];
  const float* map_ious       = (const float*)d_in[4];
  const float* pred_mask_prob = (const float*)d_in[5];
  const float* rand_vals      = (const float*)d_in[6];
  float* out     = (float*)d_out;
  float* maxprob = (float*)d_ws;   // 1024 floats of scratch

  (void)in_sizes; (void)n_in; (void)out_size; (void)ws_size;

  maxprob_kernel<<<1024, 256, 0, stream>>>(pred_mask_prob, maxprob);
  loss_kernel<<<1, 1024, 0, stream>>>(cls_logits, iou_scores, target_ids,
                                      map_indices, map_ious, rand_vals,
                                      maxprob, out);
}